// FractalEmbedding_30365418782757
// MI455X (gfx1250) — compile-verified
//
#include <hip/hip_runtime.h>

typedef __attribute__((ext_vector_type(2))) float v2f;
typedef __attribute__((ext_vector_type(8))) float v8f;

#define STEPS 8
#define NF 16               // K = 2*STEPS
#define EMBED 1024
#define NT_PER_WAVE 4       // four 16-wide N tiles => 64 dims per wave
#define WAVES_PER_BLOCK 8   // 256 threads, wave32

__global__ __launch_bounds__(256)
void fractal_embed_wmma_f32(const int* __restrict__ tok,
                            const float* __restrict__ ctab,
                            const float* __restrict__ W,       // (1024, 16) row-major
                            const float* __restrict__ scale_p,
                            float* __restrict__ out)           // (65536, 1024)
{
    const int lane = threadIdx.x & 31;
    const int wave = threadIdx.x >> 5;
    const int gw   = blockIdx.x * WAVES_PER_BLOCK + wave;
    const int NGRP = EMBED / (16 * NT_PER_WAVE);   // 16 n-groups
    const int m0   = (gw / NGRP) * 16;
    const int ngrp = gw % NGRP;

    const bool hi   = (lane >= 16);
    const int  mrow = lane & 15;                   // token row / output column in tile

    const float scale = *scale_p;

    // ---- gather c for token (m0 + mrow); both half-waves use the same token
    const int tid = tok[m0 + mrow];
    const v2f cc  = *(const v2f*)(ctab + 2 * (size_t)tid);
    const float cr = cc.x, ci = cc.y;

    // ---- 8 Julia steps, fully unrolled into named scalars (no private array!)
    float zr = 0.0f, zi = 0.0f;
    float f0,f1,f2,f3,f4,f5,f6,f7,f8,f9,f10,f11,f12,f13,f14,f15;
#define JSTEP(EVEN, ODD)                                   \
    { float nzr = zr*zr - zi*zi + cr;                      \
      zi = 2.0f*zr*zi + ci; zr = nzr;                      \
      EVEN = zr * scale; ODD = zi * scale; }
    JSTEP(f0,  f1)
    JSTEP(f2,  f3)
    JSTEP(f4,  f5)
    JSTEP(f6,  f7)
    JSTEP(f8,  f9)
    JSTEP(f10, f11)
    JSTEP(f12, f13)
    JSTEP(f14, f15)
#undef JSTEP

    // ---- A fragments for V_WMMA_F32_16X16X4_F32 (slice j covers K=4j..4j+3):
    // VGPR0 = K (lanes 0-15) / K+2 (lanes 16-31); VGPR1 = K+1 / K+3
    v2f a0 = { hi ? f2  : f0,  hi ? f3  : f1  };
    v2f a1 = { hi ? f6  : f4,  hi ? f7  : f5  };
    v2f a2 = { hi ? f10 : f8,  hi ? f11 : f9  };
    v2f a3 = { hi ? f14 : f12, hi ? f15 : f13 };

    const int khalf = hi ? 2 : 0;

#pragma unroll
    for (int nt = 0; nt < NT_PER_WAVE; ++nt) {
        const int n0 = (ngrp * NT_PER_WAVE + nt) * 16;
        const int d  = n0 + mrow;                  // this lane's output column

        // B[k][d] = W[d][k]: each lane loads exactly the 2 floats it needs per
        // slice (8B loads, immediate offsets) -- no selects, half the bytes.
        const float* wb = W + (size_t)d * NF + khalf;
        const v2f b0 = *(const v2f*)(wb + 0);      // K 4*0 + {0,1}+khalf
        const v2f b1 = *(const v2f*)(wb + 4);      // K 4*1 + ...
        const v2f b2 = *(const v2f*)(wb + 8);
        const v2f b3 = *(const v2f*)(wb + 12);

        v8f c = {};
        c = __builtin_amdgcn_wmma_f32_16x16x4_f32(false, a0, false, b0, (short)0, c, false, false);
        c = __builtin_amdgcn_wmma_f32_16x16x4_f32(false, a1, false, b1, (short)0, c, false, false);
        c = __builtin_amdgcn_wmma_f32_16x16x4_f32(false, a2, false, b2, (short)0, c, false, false);
        c = __builtin_amdgcn_wmma_f32_16x16x4_f32(false, a3, false, b3, (short)0, c, false, false);

        // D layout: VGPR r -> row M=r (lanes 0-15) / M=r+8 (lanes 16-31), N=mrow.
        // Output is a 256MB write-once stream => non-temporal stores.
        float* o = out + (size_t)(m0 + (hi ? 8 : 0)) * EMBED + n0 + mrow;
        __builtin_nontemporal_store(c[0], o + 0 * (size_t)EMBED);
        __builtin_nontemporal_store(c[1], o + 1 * (size_t)EMBED);
        __builtin_nontemporal_store(c[2], o + 2 * (size_t)EMBED);
        __builtin_nontemporal_store(c[3], o + 3 * (size_t)EMBED);
        __builtin_nontemporal_store(c[4], o + 4 * (size_t)EMBED);
        __builtin_nontemporal_store(c[5], o + 5 * (size_t)EMBED);
        __builtin_nontemporal_store(c[6], o + 6 * (size_t)EMBED);
        __builtin_nontemporal_store(c[7], o + 7 * (size_t)EMBED);
    }
}

extern "C" void kernel_launch(void* const* d_in, const int* in_sizes, int n_in,
                              void* d_out, int out_size, void* d_ws, size_t ws_size,
                              hipStream_t stream) {
    const int*   tokp  = (const int*)d_in[0];    // token_ids (B*L)
    const float* ctab  = (const float*)d_in[1];  // c_table (V,2)
    const float* W     = (const float*)d_in[2];  // (1024,16)
    const float* scale = (const float*)d_in[3];  // scalar
    float* out = (float*)d_out;

    const int n_tokens    = in_sizes[0];                            // 65536
    const int m_tiles     = n_tokens / 16;                          // 4096
    const int total_waves = m_tiles * (EMBED / (16 * NT_PER_WAVE)); // 65536
    const int blocks      = total_waves / WAVES_PER_BLOCK;          // 8192

    fractal_embed_wmma_f32<<<blocks, 256, 0, stream>>>(tokp, ctab, W, scale, out);
}